// FCFClient_58909771431936
// MI455X (gfx1250) — compile-verified
//
#include <hip/hip_runtime.h>

#define LAMBDA_REG 0.01f
constexpr int K = 256;
constexpr int I = 131072;
constexpr int N = 262144;

typedef __attribute__((ext_vector_type(2))) float v2f;
typedef __attribute__((ext_vector_type(8))) float v8f;

// ---------------------------------------------------------------------------
// Kernel 1: dot[i] = sum_k X[k] * Y[k*I + i]   (streaming GEMV, coalesced)
//           winner[i] = -1
// Y is read with default RT policy so it lands in the 192MB L2 and is still
// resident when k_grad_wmma re-reads it.
// ---------------------------------------------------------------------------
__global__ __launch_bounds__(256) void k_gemv_dots(
    const float* __restrict__ X, const float* __restrict__ Y,
    float* __restrict__ dot, int* __restrict__ winner) {
  const int i = blockIdx.x * 256 + threadIdx.x;
  float acc = 0.0f;
#pragma unroll 8
  for (int k = 0; k < K; ++k) {
    acc = fmaf(X[k], Y[(size_t)k * I + i], acc);  // X[k] uniform -> scalar path
  }
  dot[i] = acc;
  winner[i] = -1;
}

// ---------------------------------------------------------------------------
// Kernel 2: per-sample error, deterministic duplicate resolution (max n wins),
//           fixed-order block reduction of err^2 into partial sums.
// ---------------------------------------------------------------------------
__global__ __launch_bounds__(256) void k_err_winner(
    const float* __restrict__ dot, const float* __restrict__ likes,
    const int* __restrict__ ids, int* __restrict__ winner,
    float* __restrict__ partial) {
  __shared__ float sm[256];
  const int n = blockIdx.x * 256 + threadIdx.x;
  const int id = ids[n];
  const float e = dot[id] - likes[n];
  atomicMax(&winner[id], n);  // deterministic winner: largest n
  sm[threadIdx.x] = e * e;
  __syncthreads();
#pragma unroll
  for (int s = 128; s > 0; s >>= 1) {
    if (threadIdx.x < s) sm[threadIdx.x] += sm[threadIdx.x + s];
    __syncthreads();
  }
  if (threadIdx.x == 0) partial[blockIdx.x] = sm[0];
}

// ---------------------------------------------------------------------------
// Kernel 3: single-block deterministic tree reduction of 1024 partials.
// ---------------------------------------------------------------------------
__global__ __launch_bounds__(1024) void k_reduce_loss(
    const float* __restrict__ partial, float* __restrict__ out_loss) {
  __shared__ float sm[1024];
  sm[threadIdx.x] = partial[threadIdx.x];
  __syncthreads();
#pragma unroll
  for (int s = 512; s > 0; s >>= 1) {
    if (threadIdx.x < s) sm[threadIdx.x] += sm[threadIdx.x + s];
    __syncthreads();
  }
  if (threadIdx.x == 0) out_loss[0] = sm[0] * (1.0f / (float)N);
}

// ---------------------------------------------------------------------------
// Kernel 4: y_grad tiles via V_WMMA_F32_16X16X4_F32.
//   D(16x16) = A(16x4) x B(4x16) + C(16x16)
//   A col K=0 = X[k0..k0+15], B row K=0 = e[i0..i0+15], C = (2λ/N)*sel*Y tile.
// One wave per 16-column tile; iterates the 16 k-tiles (K=256).
// X staged in LDS (removes exec-masked global loads from the inner loop);
// y_grad stored non-temporally so the 128MB output stream does not evict Y
// from L2 (Y read hits from kernel 1's pass).
// ---------------------------------------------------------------------------
__global__ __launch_bounds__(256) void k_grad_wmma(
    const float* __restrict__ X, const float* __restrict__ Y,
    const float* __restrict__ dot, const float* __restrict__ likes,
    const int* __restrict__ winner, float* __restrict__ yg) {
  __shared__ float sx[K];
  sx[threadIdx.x] = X[threadIdx.x];  // blockDim == K == 256
  __syncthreads();

  const int lane = threadIdx.x & 31;
  const int wave = threadIdx.x >> 5;
  const int tile = blockIdx.x * 8 + wave;  // 8192 column tiles
  const int i0 = tile * 16;
  const int col = i0 + (lane & 15);        // this lane's column (both halves)

  const int w = winner[col];
  const bool sel = (w >= 0);
  const float e_full = sel ? ((2.0f / (float)N) * (dot[col] - likes[w])) : 0.0f;
  const float cscale = sel ? (2.0f * LAMBDA_REG / (float)N) : 0.0f;

  // B 4x16: VGPR0 lanes 0-15 hold row K=0 (N=lane); everything else zero.
  v2f b;
  b.x = (lane < 16) ? e_full : 0.0f;
  b.y = 0.0f;

  for (int kt = 0; kt < 16; ++kt) {
    const int k0 = kt * 16;
    // A 16x4: VGPR0 lanes 0-15 hold K=0 (M=lane); everything else zero.
    const float xv = sx[k0 + (lane & 15)];  // ds_load, broadcast-friendly
    v2f a;
    a.x = (lane < 16) ? xv : 0.0f;
    a.y = 0.0f;

    // C/D 16x16: VGPR r -> M = k0 + r (+8 for lanes 16-31), N = lane&15.
    const int rowBase = k0 + ((lane < 16) ? 0 : 8);
    v8f c;
#pragma unroll
    for (int r = 0; r < 8; ++r)
      c[r] = cscale * Y[(size_t)(rowBase + r) * I + col];

#if __has_builtin(__builtin_amdgcn_wmma_f32_16x16x4_f32)
    v8f d = __builtin_amdgcn_wmma_f32_16x16x4_f32(
        /*neg_a=*/false, a, /*neg_b=*/false, b,
        /*c_mod=*/(short)0, c, /*reuse_a=*/false, /*reuse_b=*/false);
#else
    // VALU fallback: D[m][n] = X[k0+m]*e_n + C[m][n]
    v8f d;
#pragma unroll
    for (int r = 0; r < 8; ++r) d[r] = fmaf(sx[rowBase + r], e_full, c[r]);
#endif

#pragma unroll
    for (int r = 0; r < 8; ++r)
      __builtin_nontemporal_store(d[r], &yg[(size_t)(rowBase + r) * I + col]);
  }
}

// ---------------------------------------------------------------------------
extern "C" void kernel_launch(void* const* d_in, const int* in_sizes, int n_in,
                              void* d_out, int out_size, void* d_ws, size_t ws_size,
                              hipStream_t stream) {
  const float* X     = (const float*)d_in[0];  // (1, 256)
  const float* Y     = (const float*)d_in[1];  // (256, 131072) row-major
  const float* likes = (const float*)d_in[2];  // (N,)
  const int*   ids   = (const int*)d_in[3];    // (N,)

  float* out_loss = (float*)d_out;        // out[0]
  float* yg       = (float*)d_out + 1;    // out[1 .. 1+K*I)

  char* ws = (char*)d_ws;
  float* dot     = (float*)ws;                      // I floats
  int*   winner  = (int*)(ws + (size_t)I * 4);      // I ints
  float* partial = (float*)(ws + (size_t)2 * I * 4);// N/256 = 1024 floats

  k_gemv_dots  <<<I / 256, 256, 0, stream>>>(X, Y, dot, winner);
  k_err_winner <<<N / 256, 256, 0, stream>>>(dot, likes, ids, winner, partial);
  k_reduce_loss<<<1, 1024, 0, stream>>>(partial, out_loss);
  k_grad_wmma  <<<(I / 16) / 8, 256, 0, stream>>>(X, Y, dot, likes, winner, yg);
}